// GCN_40948218200231
// MI455X (gfx1250) — compile-verified
//
#include <hip/hip_runtime.h>
#include <math.h>

#define G  5
#define NN 100000
#define EE 3200000
#define DD 300
#define HH 16

typedef __attribute__((ext_vector_type(2))) float v2f;
typedef __attribute__((ext_vector_type(8))) float v8f;

// ---------------------------------------------------------------------------
// 1) init: deg starts at 1.0 (self-loop), score accumulators zeroed
// ---------------------------------------------------------------------------
__global__ void k_init(float* __restrict__ deg, float* __restrict__ scores) {
  int i = blockIdx.x * blockDim.x + threadIdx.x;
  if (i < G * NN) deg[i] = 1.0f;
  if (i < 8) scores[i] = 0.0f;
}

// ---------------------------------------------------------------------------
// 2) degree accumulation over dst indices (L2-resident fp32 atomics)
// ---------------------------------------------------------------------------
__global__ void k_degree(const int* __restrict__ ei, float* __restrict__ deg) {
  int g = blockIdx.y;
  int e = blockIdx.x * blockDim.x + threadIdx.x;
  if (e >= EE) return;
  int dst = ei[(size_t)g * 2 * EE + EE + e];
  atomicAdd(&deg[(size_t)g * NN + dst], 1.0f);
}

// ---------------------------------------------------------------------------
// 3) deg -> deg^{-1/2} in place (deg >= 1 always, no zero branch needed)
// ---------------------------------------------------------------------------
__global__ void k_rsqrt(float* __restrict__ dis) {
  int i = blockIdx.x * blockDim.x + threadIdx.x;
  if (i < G * NN) dis[i] = rsqrtf(dis[i]);
}

// ---------------------------------------------------------------------------
// 4) H = X @ W1 with V_WMMA_F32_16X16X4_F32. One wave per 16-row tile,
//    K-loop of 75 steps (D=300). W1 staged in LDS (19.2 KB / block).
//    A frag: row M=lane&15, K pair = k + 2*(lane>>4) .. +1   (ISA 7.12.2)
//    B frag: col N=lane&15, same K pairing
//    C frag: VGPR r -> row r + 8*(lane>>4), col lane&15
// ---------------------------------------------------------------------------
__global__ void k_gemm_wmma(const float* __restrict__ X,
                            const float* __restrict__ W1,
                            float* __restrict__ Hout) {
  __shared__ float w[DD * HH];
  for (int i = threadIdx.x; i < DD * HH; i += blockDim.x) w[i] = W1[i];
  __syncthreads();

  const int lane = threadIdx.x & 31;
  const int wave = threadIdx.x >> 5;
  const int tile = blockIdx.x * (blockDim.x >> 5) + wave;
  const int nTiles = (G * NN) / 16;          // 500000/16 = 31250, exact
  if (tile >= nTiles) return;                // wave-uniform: EXEC stays full

  const size_t row0 = (size_t)tile * 16;
  const int m  = lane & 15;                  // A row / B col / C col
  const int kh = (lane >> 4) << 1;           // 0 or 2: K sub-offset per half

  const float* xrow = X + (row0 + m) * DD;
  v8f c = {};
#pragma unroll 5
  for (int k = 0; k < DD; k += 4) {
    v2f a, b;
    a.x = xrow[k + kh];
    a.y = xrow[k + kh + 1];
    b.x = w[(k + kh) * HH + m];
    b.y = w[(k + kh + 1) * HH + m];
    c = __builtin_amdgcn_wmma_f32_16x16x4_f32(
        /*neg_a=*/false, a, /*neg_b=*/false, b,
        /*c_mod=*/(short)0, c, /*reuse_a=*/false, /*reuse_b=*/false);
  }

  const int rbase = (lane >> 4) << 3;        // lanes 16-31 hold rows M+8
  float* orow = Hout + (row0 + rbase) * HH + m;
#pragma unroll
  for (int r = 0; r < 8; ++r) orow[(size_t)r * HH] = c[r];
}

// ---------------------------------------------------------------------------
// 5) self-loop init of layer-1 aggregate: agg1[i] = h[i] * dis[i]^2
//    (fully writes agg1, so no separate zeroing pass is needed)
// ---------------------------------------------------------------------------
__global__ void k_self1(const float* __restrict__ dis,
                        const float* __restrict__ h,
                        float* __restrict__ agg) {
  int i = blockIdx.x * blockDim.x + threadIdx.x;
  if (i >= G * NN) return;
  float s = dis[i];
  s = s * s;
  const float4* hp = (const float4*)(h + (size_t)i * HH);
  float4* ap = (float4*)(agg + (size_t)i * HH);
#pragma unroll
  for (int q = 0; q < 4; ++q) {
    float4 v = hp[q];
    v.x *= s; v.y *= s; v.z *= s; v.w *= s;
    ap[q] = v;
  }
}

// ---------------------------------------------------------------------------
// 6) layer-1 edge aggregation: agg1[dst] += h[src] * dis[src]*dis[dst]
// ---------------------------------------------------------------------------
__global__ void k_edge1(const int* __restrict__ ei,
                        const float* __restrict__ dis,
                        const float* __restrict__ h,
                        float* __restrict__ agg) {
  int g = blockIdx.y;
  int e = blockIdx.x * blockDim.x + threadIdx.x;
  if (e >= EE) return;
  const int* eig = ei + (size_t)g * 2 * EE;
  int s = eig[e];
  int d = eig[EE + e];
  const float* disg = dis + (size_t)g * NN;
  float norm = disg[s] * disg[d];
  const float4* hs = (const float4*)(h + ((size_t)g * NN + s) * HH);
  float* ad = agg + ((size_t)g * NN + d) * HH;
#pragma unroll
  for (int q = 0; q < 4; ++q) {
    float4 v = hs[q];
    atomicAdd(ad + 4 * q + 0, v.x * norm);
    atomicAdd(ad + 4 * q + 1, v.y * norm);
    atomicAdd(ad + 4 * q + 2, v.z * norm);
    atomicAdd(ad + 4 * q + 3, v.w * norm);
  }
}

// ---------------------------------------------------------------------------
// 7) layer 2 node op: h2 = relu(agg1 + b1) . W2 ; agg2 self-loop init
// ---------------------------------------------------------------------------
__global__ void k_node2(const float* __restrict__ agg1,
                        const float* __restrict__ b1,
                        const float* __restrict__ W2,
                        const float* __restrict__ dis,
                        float* __restrict__ h2,
                        float* __restrict__ agg2) {
  int i = blockIdx.x * blockDim.x + threadIdx.x;
  if (i >= G * NN) return;
  const float* a = agg1 + (size_t)i * HH;
  float acc = 0.0f;
#pragma unroll
  for (int j = 0; j < HH; ++j) acc += fmaxf(a[j] + b1[j], 0.0f) * W2[j];
  h2[i] = acc;
  float s = dis[i];
  agg2[i] = acc * s * s;
}

// ---------------------------------------------------------------------------
// 8) layer-2 edge aggregation (scalar per edge)
// ---------------------------------------------------------------------------
__global__ void k_edge2(const int* __restrict__ ei,
                        const float* __restrict__ dis,
                        const float* __restrict__ h2,
                        float* __restrict__ agg2) {
  int g = blockIdx.y;
  int e = blockIdx.x * blockDim.x + threadIdx.x;
  if (e >= EE) return;
  const int* eig = ei + (size_t)g * 2 * EE;
  int s = eig[e];
  int d = eig[EE + e];
  const float* disg = dis + (size_t)g * NN;
  float norm = disg[s] * disg[d];
  atomicAdd(&agg2[(size_t)g * NN + d], h2[(size_t)g * NN + s] * norm);
}

// ---------------------------------------------------------------------------
// 9) per-graph sum of agg2 (block-tree reduce + one atomic per block)
// ---------------------------------------------------------------------------
__global__ void k_reduce(const float* __restrict__ agg2,
                         float* __restrict__ scores) {
  __shared__ float sdata[256];
  int g = blockIdx.y;
  int tid = threadIdx.x;
  int i = blockIdx.x * blockDim.x + tid;
  sdata[tid] = (i < NN) ? agg2[(size_t)g * NN + i] : 0.0f;
  __syncthreads();
#pragma unroll
  for (int sft = 128; sft > 0; sft >>= 1) {
    if (tid < sft) sdata[tid] += sdata[tid + sft];
    __syncthreads();
  }
  if (tid == 0) atomicAdd(&scores[g], sdata[0]);
}

// ---------------------------------------------------------------------------
// 10) softmax over the 5 graph scores (mean + b2 folded in)
// ---------------------------------------------------------------------------
__global__ void k_softmax(const float* __restrict__ scores,
                          const float* __restrict__ b2,
                          float* __restrict__ out) {
  if (threadIdx.x != 0 || blockIdx.x != 0) return;
  float s[G];
  float mx = -3.402823466e+38f;
  for (int g = 0; g < G; ++g) {
    s[g] = scores[g] / (float)NN + b2[0];
    mx = fmaxf(mx, s[g]);
  }
  float sum = 0.0f;
  for (int g = 0; g < G; ++g) {
    s[g] = __expf(s[g] - mx);
    sum += s[g];
  }
  for (int g = 0; g < G; ++g) out[g] = s[g] / sum;
}

// ---------------------------------------------------------------------------
extern "C" void kernel_launch(void* const* d_in, const int* in_sizes, int n_in,
                              void* d_out, int out_size, void* d_ws, size_t ws_size,
                              hipStream_t stream) {
  const float* emb = (const float*)d_in[0];   // [5, N, 300]
  const int*   ei  = (const int*)d_in[1];     // [5, 2, E]
  const float* W1  = (const float*)d_in[2];   // [300, 16]
  const float* b1  = (const float*)d_in[3];   // [16]
  const float* W2  = (const float*)d_in[4];   // [16, 1]
  const float* b2  = (const float*)d_in[5];   // [1]
  float* out = (float*)d_out;                 // [5]

  // workspace layout (floats): dis | h | agg1 | h2 | agg2 | scores
  float* ws     = (float*)d_ws;
  float* dis    = ws;                                   // G*NN
  float* h      = dis  + (size_t)G * NN;                // G*NN*HH
  float* agg1   = h    + (size_t)G * NN * HH;           // G*NN*HH
  float* h2     = agg1 + (size_t)G * NN * HH;           // G*NN
  float* agg2   = h2   + (size_t)G * NN;                // G*NN
  float* scores = agg2 + (size_t)G * NN;                // 8

  const int T = 256;
  const int nodeBlocks = (G * NN + T - 1) / T;          // 1954
  const dim3 edgeGrid((EE + T - 1) / T, G);             // (12500, 5)
  const int gemmTiles  = (G * NN) / 16;                 // 31250
  const int gemmBlocks = (gemmTiles + (T / 32) - 1) / (T / 32);

  k_init   <<<nodeBlocks, T, 0, stream>>>(dis, scores);
  k_degree <<<edgeGrid,  T, 0, stream>>>(ei, dis);
  k_rsqrt  <<<nodeBlocks, T, 0, stream>>>(dis);
  k_gemm_wmma<<<gemmBlocks, T, 0, stream>>>(emb, W1, h);
  k_self1  <<<nodeBlocks, T, 0, stream>>>(dis, h, agg1);
  k_edge1  <<<edgeGrid,  T, 0, stream>>>(ei, dis, h, agg1);
  k_node2  <<<nodeBlocks, T, 0, stream>>>(agg1, b1, W2, dis, h2, agg2);
  k_edge2  <<<edgeGrid,  T, 0, stream>>>(ei, dis, h2, agg2);
  k_reduce <<<dim3((NN + T - 1) / T, G), T, 0, stream>>>(agg2, scores);
  k_softmax<<<1, 32, 0, stream>>>(scores, b2, out);
}